// UNISAGEModel_63720134804239
// MI455X (gfx1250) — compile-verified
//
#include <hip/hip_runtime.h>
#include <hip/hip_bf16.h>

#define HD 128               // hidden dim
#define DIN 14               // one-hot input dim

typedef __attribute__((ext_vector_type(2))) float v2f;
typedef __attribute__((ext_vector_type(8))) float v8f;

// ---------------------------------------------------------------- utilities
__global__ void zero_f(float* __restrict__ p, size_t n) {
    size_t i = (size_t)blockIdx.x * blockDim.x + threadIdx.x;
    if (i < n) p[i] = 0.0f;
}

// deg[v] += 1 per incidence nonzero
__global__ void deg_count(const int* __restrict__ rows, float* __restrict__ deg, int nnz) {
    int k = blockIdx.x * blockDim.x + threadIdx.x;
    if (k < nnz) unsafeAtomicAdd(&deg[rows[k]], 1.0f);
}

// deg -> 1/max(deg,1) in place
__global__ void inv_deg_k(float* __restrict__ deg, int n) {
    int v = blockIdx.x * blockDim.x + threadIdx.x;
    if (v < n) deg[v] = 1.0f / fmaxf(deg[v], 1.0f);
}

// ------------------------------------------------- input projection (K=14)
__global__ __launch_bounds__(HD) void in_proj(const float* __restrict__ X,
                                              const float* __restrict__ W,
                                              const float* __restrict__ b,
                                              float* __restrict__ Y, int M) {
    __shared__ float sx[DIN];
    int v = blockIdx.x;
    if (v >= M) return;
    if (threadIdx.x < DIN) sx[threadIdx.x] = X[(size_t)v * DIN + threadIdx.x];
    __syncthreads();
    int j = threadIdx.x;
    float acc = b[j];
#pragma unroll
    for (int k = 0; k < DIN; ++k) acc += sx[k] * W[k * HD + j];
    Y[(size_t)v * HD + j] = acc;
}

// ------------------------------------------ fp32 WMMA GEMM: Y = X@W + bias
// X:[M,128] row-major, W:[128,128] row-major, M multiple of 16.
// Block = 256 threads = 8 waves; each wave -> one 16-row strip x 128 cols.
//
// W is staged in LDS pair-interleaved + XOR-swizzled:
//   element (k,n) -> word (k>>1)*256 + ((n*2) ^ (((k>>1)&1)<<5)) + (k&1)
// so a B fragment (rows k,k+1 at column n) is one aligned ds_load_b64, and the
// two half-waves (even vs odd pair-rows) touch disjoint bank halves.
__global__ __launch_bounds__(256) void gemm_wmma_f32(const float* __restrict__ X,
                                                     const float* __restrict__ W,
                                                     const float* __restrict__ bias,
                                                     float* __restrict__ Y, int M) {
    __shared__ float sW[HD * HD];  // 64 KB
    for (int i = threadIdx.x; i < HD * HD; i += blockDim.x) {
        int k = i >> 7, n = i & (HD - 1);
        int pr = k >> 1;
        sW[pr * 256 + (((n << 1) ^ ((pr & 1) << 5))) + (k & 1)] = W[i];
    }
    __syncthreads();

    const int wave  = threadIdx.x >> 5;
    const int lane  = threadIdx.x & 31;
    const int nlo   = lane & 15;         // N (and M-row) index within half-wave
    const int khalf = (lane >> 4) * 2;   // A/B K sub-offset: 0 or 2
    const int prOff = khalf >> 1;        // pair-row offset: 0 or 1
    const int xorv  = prOff << 5;        // lane-constant swizzle term

    int numTiles = M >> 4;               // M % 16 == 0 (100000/16 = 6250)
    int tile = blockIdx.x * 8 + wave;
    if (tile >= numTiles) tile = numTiles - 1;  // clamped waves redo last tile (same values)
    const int m0 = tile << 4;

    // accumulators: acc[nt] covers cols [nt*16, nt*16+16)
    v8f acc[8];
#pragma unroll
    for (int nt = 0; nt < 8; ++nt) {
        float bv = bias[nt * 16 + nlo];
#pragma unroll
        for (int r = 0; r < 8; ++r) acc[nt][r] = bv;
    }

    // A fragment per ISA: lane L -> M = L%16; VGPR0 = K + (L<16?0:2), VGPR1 = +1
    const float* arow = X + (size_t)(m0 + nlo) * HD;
    for (int k0 = 0; k0 < HD; k0 += 4) {
        v2f a;
        a.x = arow[k0 + khalf];
        a.y = arow[k0 + khalf + 1];
        const int prBase = ((k0 >> 1) + prOff) * 256;
#pragma unroll
        for (int nt = 0; nt < 8; ++nt) {
            const int n2 = ((nt * 16 + nlo) << 1) ^ xorv;
            v2f b = *(const v2f*)&sW[prBase + n2];   // ds_load_b64, conflict-free
            acc[nt] = __builtin_amdgcn_wmma_f32_16x16x4_f32(
                false, a, false, b, (short)0, acc[nt], false, false);
        }
    }

    // C/D layout: VGPR r, lanes 0-15 -> M = r, lanes 16-31 -> M = r+8; N = lane%16
    const int rowoff = (lane >> 4) * 8;
#pragma unroll
    for (int nt = 0; nt < 8; ++nt) {
#pragma unroll
        for (int r = 0; r < 8; ++r) {
            Y[(size_t)(m0 + rowoff + r) * HD + nt * 16 + nlo] = acc[nt][r];
        }
    }
}

// ------------------------------------------------- scatter kernels (1 wave / nnz)
// e[cols[k]] += h[rows[k]]
__global__ __launch_bounds__(256) void scatter_h_to_e(const float* __restrict__ h,
                                                      float* __restrict__ e,
                                                      const int* __restrict__ rows,
                                                      const int* __restrict__ cols, int nnz) {
    int w = blockIdx.x * (blockDim.x >> 5) + (threadIdx.x >> 5);
    int lane = threadIdx.x & 31;
    if (w >= nnz) return;
    int r = rows[w], c = cols[w];
    float4 v = ((const float4*)(h + (size_t)r * HD))[lane];
    float* dst = e + (size_t)c * HD + lane * 4;
    unsafeAtomicAdd(dst + 0, v.x);
    unsafeAtomicAdd(dst + 1, v.y);
    unsafeAtomicAdd(dst + 2, v.z);
    unsafeAtomicAdd(dst + 3, v.w);
}

// x[rows[k]] += e[cols[k]] * inv_deg[rows[k]]   (x holds h; becomes new x0)
__global__ __launch_bounds__(256) void scatter_e_to_x(const float* __restrict__ e,
                                                      float* __restrict__ x,
                                                      const float* __restrict__ invdeg,
                                                      const int* __restrict__ rows,
                                                      const int* __restrict__ cols, int nnz) {
    int w = blockIdx.x * (blockDim.x >> 5) + (threadIdx.x >> 5);
    int lane = threadIdx.x & 31;
    if (w >= nnz) return;
    int r = rows[w], c = cols[w];
    float s = invdeg[r];
    float4 v = ((const float4*)(e + (size_t)c * HD))[lane];
    float* dst = x + (size_t)r * HD + lane * 4;
    unsafeAtomicAdd(dst + 0, v.x * s);
    unsafeAtomicAdd(dst + 1, v.y * s);
    unsafeAtomicAdd(dst + 2, v.z * s);
    unsafeAtomicAdd(dst + 3, v.w * s);
}

// ------------------------------------------------- column sums: out[128] += sum rows
__global__ __launch_bounds__(256) void colsum_k(const float* __restrict__ X,
                                                float* __restrict__ out, int M) {
    int j = threadIdx.x & (HD - 1);
    int sub = threadIdx.x >> 7;  // 0/1
    int rows_per = (M + gridDim.x - 1) / gridDim.x;
    int r0 = blockIdx.x * rows_per;
    int r1 = min(r0 + rows_per, M);
    float acc = 0.0f;
    for (int r = r0 + sub; r < r1; r += 2) acc += X[(size_t)r * HD + j];
    unsafeAtomicAdd(&out[j], acc);
}

// out[0] = cs0.Wo0/N_V + bo0 + cs1.Wo1/N_E + bo1
__global__ __launch_bounds__(HD) void final_k(const float* __restrict__ cs0,
                                              const float* __restrict__ cs1,
                                              const float* __restrict__ Wo0,
                                              const float* __restrict__ bo0,
                                              const float* __restrict__ Wo1,
                                              const float* __restrict__ bo1,
                                              float* __restrict__ out, int NV, int NE) {
    __shared__ float red[HD];
    int j = threadIdx.x;
    red[j] = cs0[j] * Wo0[j] / (float)NV + cs1[j] * Wo1[j] / (float)NE;
    __syncthreads();
    for (int s = HD / 2; s > 0; s >>= 1) {
        if (j < s) red[j] += red[j + s];
        __syncthreads();
    }
    if (j == 0) out[0] = red[0] + bo0[0] + bo1[0];
}

// ---------------------------------------------------------------- launcher
extern "C" void kernel_launch(void* const* d_in, const int* in_sizes, int n_in,
                              void* d_out, int out_size, void* d_ws, size_t ws_size,
                              hipStream_t stream) {
    (void)n_in; (void)out_size; (void)ws_size;

    const float* x_0  = (const float*)d_in[0];
    // d_in[1] = x_1 : dead in the reference (projection overwritten) -> skipped
    const int*   rows = (const int*)d_in[2];
    const int*   cols = (const int*)d_in[3];
    const float* W0i  = (const float*)d_in[4];
    const float* b0i  = (const float*)d_in[5];
    const float* Wl0  = (const float*)d_in[8];
    const float* bl0  = (const float*)d_in[9];
    const float* Wl1  = (const float*)d_in[10];
    const float* bl1  = (const float*)d_in[11];
    const float* Wo0  = (const float*)d_in[12];
    const float* bo0  = (const float*)d_in[13];
    const float* Wo1  = (const float*)d_in[14];
    const float* bo1  = (const float*)d_in[15];
    float* out = (float*)d_out;

    const int N_V = in_sizes[0] / DIN;   // 100000
    const int N_E = in_sizes[1] / DIN;   // 200000
    const int NNZ = in_sizes[2];         // 800000

    const size_t NVH = (size_t)N_V * HD;
    const size_t NEH = (size_t)N_E * HD;

    float* bufA = (float*)d_ws;          // x0 / layer2 output      (51.2 MB)
    float* bufB = bufA + NVH;            // h / layer1 output       (51.2 MB)
    float* eBuf = bufB + NVH;            // hyperedge features      (102.4 MB)
    float* deg  = eBuf + NEH;            // degrees -> inv_deg      (0.4 MB)
    float* cs0  = deg + N_V;             // colsum(x0) [128]
    float* cs1  = cs0 + HD;              // colsum(e)  [128]

    const int TPB = 256;
    const int gemmBlocks = ((N_V >> 4) + 7) / 8;      // 8 tiles per block
    const int scatBlocks = (NNZ + 7) / 8;             // 1 wave per nnz

    // degrees
    zero_f<<<(N_V + TPB - 1) / TPB, TPB, 0, stream>>>(deg, (size_t)N_V);
    deg_count<<<(NNZ + TPB - 1) / TPB, TPB, 0, stream>>>(rows, deg, NNZ);
    inv_deg_k<<<(N_V + TPB - 1) / TPB, TPB, 0, stream>>>(deg, N_V);

    // x0 = x_0 @ W0_in + b0_in
    in_proj<<<N_V, HD, 0, stream>>>(x_0, W0i, b0i, bufA, N_V);

    // ---- layer 1: h = x0@Wl0+bl0 ; e = B^T h ; x0 = h + (B e)/deg
    gemm_wmma_f32<<<gemmBlocks, TPB, 0, stream>>>(bufA, Wl0, bl0, bufB, N_V);
    zero_f<<<(int)((NEH + TPB - 1) / TPB), TPB, 0, stream>>>(eBuf, NEH);
    scatter_h_to_e<<<scatBlocks, TPB, 0, stream>>>(bufB, eBuf, rows, cols, NNZ);
    scatter_e_to_x<<<scatBlocks, TPB, 0, stream>>>(eBuf, bufB, deg, rows, cols, NNZ);

    // ---- layer 2 (bufB is new x0)
    gemm_wmma_f32<<<gemmBlocks, TPB, 0, stream>>>(bufB, Wl1, bl1, bufA, N_V);
    zero_f<<<(int)((NEH + TPB - 1) / TPB), TPB, 0, stream>>>(eBuf, NEH);
    scatter_h_to_e<<<scatBlocks, TPB, 0, stream>>>(bufA, eBuf, rows, cols, NNZ);
    scatter_e_to_x<<<scatBlocks, TPB, 0, stream>>>(eBuf, bufA, deg, rows, cols, NNZ);

    // ---- readout: means via column sums
    zero_f<<<1, TPB, 0, stream>>>(cs0, (size_t)(2 * HD));
    colsum_k<<<1024, TPB, 0, stream>>>(bufA, cs0, N_V);
    colsum_k<<<1024, TPB, 0, stream>>>(eBuf, cs1, N_E);
    final_k<<<1, HD, 0, stream>>>(cs0, cs1, Wo0, bo0, Wo1, bo1, out, N_V, N_E);
}